// DGCNN_encoder_34213709480364
// MI455X (gfx1250) — compile-verified
//
#include <hip/hip_runtime.h>

#define KNN 27
#define EPSV 1e-5f

typedef float v2f __attribute__((ext_vector_type(2)));
typedef float v8f __attribute__((ext_vector_type(8)));

// ---------------------------------------------------------------- prep kernels
__global__ void pad_x_kernel(const float* __restrict__ x, float* __restrict__ F0, int BN) {
  int t = blockIdx.x * blockDim.x + threadIdx.x;
  if (t < BN * 4) { int row = t >> 2, j = t & 3; F0[t] = (j < 3) ? x[row * 3 + j] : 0.f; }
}

__global__ void xx_kernel(const float* __restrict__ x, float* __restrict__ xx, int BN) {
  int t = blockIdx.x * blockDim.x + threadIdx.x;
  if (t < BN) {
    float a = x[3 * t], b = x[3 * t + 1], c = x[3 * t + 2];
    xx[t] = a * a + b * b + c * c;
  }
}

// streaming top-27 (largest d = 2<x,y> - |x|^2 - |y|^2) with replace-min
__global__ void knn_kernel(const float* __restrict__ x, const float* __restrict__ xx,
                           int* __restrict__ idx, int BN, int N) {
  int t = blockIdx.x * blockDim.x + threadIdx.x;
  if (t >= BN) return;
  int b = t / N, n = t - b * N;
  const float* xb  = x  + (size_t)b * N * 3;
  const float* xxb = xx + (size_t)b * N;
  float x0 = xb[3 * n], x1 = xb[3 * n + 1], x2 = xb[3 * n + 2], xn = xxb[n];
  float dval[KNN]; int dind[KNN];
#pragma unroll
  for (int j = 0; j < KNN; j++) { dval[j] = -3.0e38f; dind[j] = 0; }
  float worst = -3.0e38f; int wpos = 0;
  for (int m = 0; m < N; m++) {
    float d = 2.f * (x0 * xb[3 * m] + x1 * xb[3 * m + 1] + x2 * xb[3 * m + 2]) - xn - xxb[m];
    if (d > worst) {
      dval[wpos] = d; dind[wpos] = m;
      float w = dval[0]; int wp = 0;
#pragma unroll
      for (int j = 1; j < KNN; j++) { if (dval[j] < w) { w = dval[j]; wp = j; } }
      worst = w; wpos = wp;
    }
  }
#pragma unroll
  for (int j = 0; j < KNN; j++) idx[(size_t)t * KNN + j] = dind[j];
}

// Packed pair-row layout for B matrices: Bpack[((c>>1)*Cout + o)*2 + (c&1)] = B[c][o].
// Makes a WMMA B-fragment (rows {k,k+1}, one column) 8 contiguous bytes -> one b64 load.
__global__ void prep_w_kernel(const float* __restrict__ W, float* __restrict__ WdP,
                              float* __restrict__ WmP, int Cin, int Cout) {
  int o = blockIdx.x * blockDim.x + threadIdx.x;
  int c = blockIdx.y;  // 0..Kpad-1
  if (o >= Cout) return;
  float wd = 0.f, wm = 0.f;
  if (c < Cin) {
    wd = W[(size_t)o * 2 * Cin + c];
    wm = W[(size_t)o * 2 * Cin + Cin + c] - wd;
  }
  size_t p = (((size_t)(c >> 1) * Cout) + o) * 2 + (c & 1);
  WdP[p] = wd;
  WmP[p] = wm;
}

__global__ void prep_w4t_kernel(const float* __restrict__ W4, float* __restrict__ W4P) {
  int o = threadIdx.x;   // 256
  int c = blockIdx.y;    // 1440
  W4P[(((size_t)(c >> 1) * 256) + o) * 2 + (c & 1)] = W4[(size_t)o * 1440 + c];
}

__global__ void zero_kernel(float* p, int n) {
  int t = blockIdx.x * blockDim.x + threadIdx.x;
  if (t < n) p[t] = 0.f;
}

// ---------------------------------------------------------------- WMMA fp32 GEMM
// O1 = F(M x Kd, ld=ldf) @ B1(Kd x Nc, pair-packed, ldb=Nc); optionally O2 = F @ B2
// (A-fragment reuse). One wave per 16x16 output tile. Fragment layouts per ISA 7.12.2:
//   A 16x4  : lanes 0-15 rows M=lane, K={0,1} in v0/v1; lanes 16-31 K={2,3}
//   B 4x16  : v0/v1 = rows K={khalf,khalf+1}, col N=lane&15  (pair index p=k0/2+(lane>=16))
//   C/D     : VGPR j -> (M=j, N=lane) lanes 0-15 ; (M=j+8, N=lane-16) lanes 16-31
template <bool DUAL>
__global__ void gemm_f32_wmma(const float* __restrict__ F, int ldf,
                              const float* __restrict__ B1, const float* __restrict__ B2, int ldb,
                              float* __restrict__ O1, float* __restrict__ O2, int ldo,
                              int M, int Kd, int Nc) {
  int gw   = (blockIdx.x * blockDim.x + threadIdx.x) >> 5;
  int lane = threadIdx.x & 31;
  int ntiles = Nc >> 4;
  int mt = gw / ntiles, nt = gw - mt * ntiles;
  if (mt * 16 >= M) return;
  int row0 = mt << 4, col0 = nt << 4;
  int lrow  = lane & 15;
  int khalf = (lane >> 4) << 1;

  const float* fr  = F + (size_t)(row0 + lrow) * ldf + khalf;
  // packed B: element pair at ((p*ldb)+col)*2, p = k0/2 + (lane>=16)
  const float* bc1 = B1 + (((size_t)(lane >> 4) * ldb) + col0 + lrow) * 2;
  const float* bc2 = DUAL ? (B2 + (((size_t)(lane >> 4) * ldb) + col0 + lrow) * 2) : nullptr;

  v8f acc1 = {}; v8f acc2 = {};
  for (int k0 = 0; k0 < Kd; k0 += 4) {
    __builtin_prefetch(fr + 64, 0, 1);            // global_prefetch_b8 on A stream
    v2f a  = *(const v2f*)fr;                     // global_load_b64
    v2f b1 = *(const v2f*)bc1;                    // global_load_b64
    acc1 = __builtin_amdgcn_wmma_f32_16x16x4_f32(false, a, false, b1, (short)0, acc1, false, false);
    if (DUAL) {
      v2f b2 = *(const v2f*)bc2;                  // global_load_b64
      acc2 = __builtin_amdgcn_wmma_f32_16x16x4_f32(false, a, false, b2, (short)0, acc2, false, false);
    }
    fr  += 4;
    bc1 += 4 * (size_t)ldb;                       // 2 pair-rows * ldb * 2 floats
    if (DUAL) bc2 += 4 * (size_t)ldb;
  }
  int orow = row0 + ((lane >> 4) << 3);
  int ocol = col0 + lrow;
#pragma unroll
  for (int j = 0; j < 8; j++) {
    O1[(size_t)(orow + j) * ldo + ocol] = acc1[j];
    if (DUAL) O2[(size_t)(orow + j) * ldo + ocol] = acc2[j];
  }
}

// ---------------------------------------------------------------- BN statistics
// per-channel sum & sumsq of y[row,k,o] = A[gather,o] + C[row,o] over all (row,k)
__global__ void edge_stats_kernel(const float* __restrict__ A, const float* __restrict__ C,
                                  const int* __restrict__ idx, float* __restrict__ stats,
                                  int Cout, int BN, int N) {
  __shared__ float rs[8 * 32], rq[8 * 32];
  int lane = threadIdx.x & 31;
  int sub  = threadIdx.x >> 5;
  int o = blockIdx.x * 32 + lane;
  float s = 0.f, q = 0.f;
  for (int row = blockIdx.y * 8 + sub; row < BN; row += gridDim.y * 8) {
    int b = row / N;
    float cen = C[(size_t)row * Cout + o];
    const int* ip = idx + (size_t)row * KNN;
#pragma unroll
    for (int k = 0; k < KNN; k++) {
      int m = ip[k];
      float y = A[(size_t)(b * N + m) * Cout + o] + cen;
      s += y; q += y * y;
    }
  }
  rs[sub * 32 + lane] = s; rq[sub * 32 + lane] = q;
  __syncthreads();
  if (threadIdx.x < 32) {
    float ts = 0.f, tq = 0.f;
#pragma unroll
    for (int j = 0; j < 8; j++) { ts += rs[j * 32 + threadIdx.x]; tq += rq[j * 32 + threadIdx.x]; }
    atomicAdd(&stats[o], ts);
    atomicAdd(&stats[Cout + o], tq);
  }
}

__global__ void row_stats_kernel(const float* __restrict__ Y, float* __restrict__ stats,
                                 int Cout, int BN) {
  __shared__ float rs[8 * 32], rq[8 * 32];
  int lane = threadIdx.x & 31;
  int sub  = threadIdx.x >> 5;
  int o = blockIdx.x * 32 + lane;
  float s = 0.f, q = 0.f;
  for (int row = blockIdx.y * 8 + sub; row < BN; row += gridDim.y * 8) {
    float y = Y[(size_t)row * Cout + o];
    s += y; q += y * y;
  }
  rs[sub * 32 + lane] = s; rq[sub * 32 + lane] = q;
  __syncthreads();
  if (threadIdx.x < 32) {
    float ts = 0.f, tq = 0.f;
#pragma unroll
    for (int j = 0; j < 8; j++) { ts += rs[j * 32 + threadIdx.x]; tq += rq[j * 32 + threadIdx.x]; }
    atomicAdd(&stats[o], ts);
    atomicAdd(&stats[Cout + o], tq);
  }
}

__global__ void finalize_stats_kernel(const float* __restrict__ stats, const float* __restrict__ g,
                                      const float* __restrict__ beta, float* __restrict__ ss,
                                      int Cout, float invS) {
  int o = blockIdx.x * blockDim.x + threadIdx.x;
  if (o >= Cout) return;
  float m   = stats[o] * invS;
  float v   = stats[Cout + o] * invS - m * m;
  float inv = rsqrtf(v + EPSV);
  float sc  = inv * g[o];
  ss[o]        = sc;
  ss[Cout + o] = beta[o] - m * sc;
}

// ---------------------------------------------------------------- activation+max
__global__ void edge_max_kernel(const float* __restrict__ A, const float* __restrict__ C,
                                const int* __restrict__ idx, const float* __restrict__ ss,
                                float* __restrict__ H, int ldH, int coff, int Cout, int N) {
  int o = blockIdx.x * blockDim.x + threadIdx.x;
  if (o >= Cout) return;
  int row = blockIdx.y;
  int b = row / N;
  float cen = C[(size_t)row * Cout + o];
  float sc = ss[o], sh = ss[Cout + o];
  const int* ip = idx + (size_t)row * KNN;
  float best = -3.0e38f;
#pragma unroll
  for (int k = 0; k < KNN; k++) {
    int m = ip[k];
    float y = (A[(size_t)(b * N + m) * Cout + o] + cen) * sc + sh;
    y = (y >= 0.f) ? y : 0.2f * y;
    best = fmaxf(best, y);
  }
  H[(size_t)row * ldH + coff + o] = best;
}

__global__ void bn_act_kernel(const float* __restrict__ Y, const float* __restrict__ ss,
                              float* __restrict__ out, int Cout) {
  int o = blockIdx.x * blockDim.x + threadIdx.x;
  if (o >= Cout) return;
  int row = blockIdx.y;
  float y = Y[(size_t)row * Cout + o] * ss[o] + ss[Cout + o];
  out[(size_t)row * Cout + o] = (y >= 0.f) ? y : 0.2f * y;
}

__global__ void gmax_kernel(const float* __restrict__ feats, float* __restrict__ gout,
                            int N, int Cout) {
  int o = blockIdx.x * blockDim.x + threadIdx.x;
  if (o >= Cout) return;
  int b = blockIdx.y;
  const float* fp = feats + (size_t)b * N * Cout + o;
  float best = -3.0e38f;
  for (int n = 0; n < N; n++) best = fmaxf(best, fp[(size_t)n * Cout]);
  gout[(size_t)b * Cout + o] = best;
}

// ---------------------------------------------------------------- driver
extern "C" void kernel_launch(void* const* d_in, const int* in_sizes, int n_in,
                              void* d_out, int out_size, void* d_ws, size_t ws_size,
                              hipStream_t stream) {
  (void)in_sizes; (void)n_in; (void)out_size; (void)ws_size;
  const int B = 2, N = 2048, BN = B * N;
  const float* x  = (const float*)d_in[0];
  const float* Wl[4] = {(const float*)d_in[1], (const float*)d_in[4], (const float*)d_in[7], (const float*)d_in[10]};
  const float* gl[4] = {(const float*)d_in[2], (const float*)d_in[5], (const float*)d_in[8], (const float*)d_in[11]};
  const float* bl[4] = {(const float*)d_in[3], (const float*)d_in[6], (const float*)d_in[9], (const float*)d_in[12]};
  const float* W4 = (const float*)d_in[13];
  const float* g4 = (const float*)d_in[14];
  const float* b4 = (const float*)d_in[15];
  float* out   = (float*)d_out;
  float* feats = out + (size_t)B * 256;

  char* ws = (char*)d_ws; size_t off = 0;
  auto take = [&](size_t bytes) -> char* {
    char* r = ws + off; off = (off + bytes + 255) & ~(size_t)255; return r;
  };
  int*   idx  = (int*)  take((size_t)BN * KNN * sizeof(int));
  float* xx   = (float*)take((size_t)BN * sizeof(float));
  float* F0   = (float*)take((size_t)BN * 4 * sizeof(float));
  float* H    = (float*)take((size_t)BN * 1440 * sizeof(float));
  float* Abuf = (float*)take((size_t)BN * 768 * sizeof(float));
  float* Cbuf = (float*)take((size_t)BN * 768 * sizeof(float));
  float* Yraw = (float*)take((size_t)BN * 256 * sizeof(float));
  float* WdP  = (float*)take((size_t)384 * 768 * sizeof(float));
  float* WmP  = (float*)take((size_t)384 * 768 * sizeof(float));
  float* W4P  = (float*)take((size_t)1440 * 256 * sizeof(float));
  float* stats= (float*)take((size_t)2 * 768 * sizeof(float));
  float* ss   = (float*)take((size_t)2 * 768 * sizeof(float));

  pad_x_kernel<<<(BN * 4 + 255) / 256, 256, 0, stream>>>(x, F0, BN);
  xx_kernel<<<(BN + 255) / 256, 256, 0, stream>>>(x, xx, BN);
  knn_kernel<<<(BN + 127) / 128, 128, 0, stream>>>(x, xx, idx, BN, N);

  struct Lcfg { int Cin, Kpad, Cout, coff; };
  const Lcfg L[4] = {{3, 4, 96, 0}, {96, 96, 192, 96}, {192, 192, 384, 288}, {384, 384, 768, 672}};
  const float* Fin = F0; int ldf = 4;
  for (int l = 0; l < 4; l++) {
    int Cin = L[l].Cin, Kpad = L[l].Kpad, Cout = L[l].Cout, coff = L[l].coff;
    prep_w_kernel<<<dim3((Cout + 255) / 256, Kpad), 256, 0, stream>>>(Wl[l], WdP, WmP, Cin, Cout);
    int waves = (BN / 16) * (Cout / 16);
    gemm_f32_wmma<true><<<(waves * 32 + 255) / 256, 256, 0, stream>>>(
        Fin, ldf, WdP, WmP, Cout, Abuf, Cbuf, Cout, BN, Kpad, Cout);
    zero_kernel<<<(2 * Cout + 255) / 256, 256, 0, stream>>>(stats, 2 * Cout);
    edge_stats_kernel<<<dim3(Cout / 32, 64), 256, 0, stream>>>(Abuf, Cbuf, idx, stats, Cout, BN, N);
    finalize_stats_kernel<<<(Cout + 255) / 256, 256, 0, stream>>>(
        stats, gl[l], bl[l], ss, Cout, 1.0f / ((float)BN * KNN));
    edge_max_kernel<<<dim3((Cout + 255) / 256, BN), 256, 0, stream>>>(
        Abuf, Cbuf, idx, ss, H, 1440, coff, Cout, N);
    Fin = H + coff; ldf = 1440;
  }

  prep_w4t_kernel<<<dim3(1, 1440), 256, 0, stream>>>(W4, W4P);
  int waves = (BN / 16) * (256 / 16);
  gemm_f32_wmma<false><<<(waves * 32 + 255) / 256, 256, 0, stream>>>(
      H, 1440, W4P, nullptr, 256, Yraw, nullptr, 256, BN, 1440, 256);
  zero_kernel<<<2, 256, 0, stream>>>(stats, 512);
  row_stats_kernel<<<dim3(256 / 32, 64), 256, 0, stream>>>(Yraw, stats, 256, BN);
  finalize_stats_kernel<<<1, 256, 0, stream>>>(stats, g4, b4, ss, 256, 1.0f / (float)BN);
  bn_act_kernel<<<dim3(1, BN), 256, 0, stream>>>(Yraw, ss, feats, 256);
  gmax_kernel<<<dim3(1, B), 256, 0, stream>>>(feats, out, N, 256);
}